// ModelWithFilterDet_62629213110364
// MI455X (gfx1250) — compile-verified
//
#include <hip/hip_runtime.h>
#include <float.h>

// Problem constants (match reference)
#define Bn      8
#define Nn      49104
#define Cn      30
#define MAXDET  100
// SCORE_THRESH = 0.01, NMS_THRESH = 0.5, handled inline.

typedef unsigned int v4u __attribute__((ext_vector_type(4)));
typedef int          v8i __attribute__((ext_vector_type(8)));
typedef int          v4i __attribute__((ext_vector_type(4)));

// ---------------------------------------------------------------------------
// Kernel 1: one workgroup per (batch, class). Scores live in LDS (Nn floats =
// 196,416 B of the WGP's 320 KB LDS). Staged via the Tensor Data Mover:
// one D# describing a 1 x N tile with tensor_dim0_stride = C gathers the
// 120-byte-strided class column into contiguous LDS in a single instruction.
// ---------------------------------------------------------------------------
__global__ __launch_bounds__(1024, 1)
void nms_kernel(const float* __restrict__ boxes,   // [B,N,4]
                const float* __restrict__ cls,     // [B,N,C]
                float* __restrict__ ws_sc,         // [B*C, MAXDET]
                int*   __restrict__ ws_ix) {       // [B*C, MAXDET]
  extern __shared__ float sm[];                    // Nn floats: class scores
  __shared__ float red_v[32];
  __shared__ int   red_i[32];
  __shared__ float s_best_v;
  __shared__ int   s_best_i;
  __shared__ float s_box[4];
  __shared__ float s_area;

  const int bc   = blockIdx.x;
  const int b    = bc / Cn;
  const int c    = bc % Cn;
  const int tid  = threadIdx.x;
  const int bs   = blockDim.x;       // 1024 = 32 waves (wave32)
  const int lane = tid & 31;
  const int wid  = tid >> 5;

#if __has_builtin(__builtin_amdgcn_tensor_load_to_lds)
  // --- TDM strided gather: cls[b, 0..N-1, c] -> sm[0..N-1] -----------------
  if (wid == 0) {
    unsigned long long ga =
        (unsigned long long)(const void*)(cls + ((size_t)b * Nn * Cn + c));
    // Low 32 bits of the flat shared-aperture address == wave LDS byte offset
    unsigned lds_off = (unsigned)(unsigned long long)(const void*)&sm[0];

    // D# group 0 (CDNA5 ISA): count=1, lds_addr,
    // global_addr[56:0], type=2 ("image") in bits 127:126.
    v4u g0;
    g0[0] = 1u;
    g0[1] = lds_off;
    g0[2] = (unsigned)(ga & 0xFFFFFFFFu);
    g0[3] = (unsigned)((ga >> 32) & 0x01FFFFFFu) | (2u << 30);

    // D# group 1 (§8.4): data_size=4B, tensor_dim0=C, tensor_dim1=N,
    // tile_dim0=1, tile_dim1=N, tensor_dim0_stride=C. No padding/multicast.
    v8i g1;
    g1[0] = (int)(2u << 16);                            // data_size = 2 (4B)
    g1[1] = (int)(((unsigned)Cn & 0xFFFFu) << 16);      // tensor_dim0 lo16
    g1[2] = (int)(((unsigned)Nn & 0xFFFFu) << 16);      // dim0 hi16=0 | dim1 lo16
    g1[3] = (int)(1u << 16);                            // dim1 hi16=0 | tile_dim0=1
    g1[4] = (int)((unsigned)Nn & 0xFFFFu);              // tile_dim1=N | tile_dim2=0
    g1[5] = (int)Cn;                                    // tensor_dim0_stride lo32
    g1[6] = 0;                                          // stride hi | dim1_stride lo
    g1[7] = 0;                                          // dim1_stride hi
    v4i gz = {0, 0, 0, 0};
#if defined(__clang_major__) && (__clang_major__ >= 23)
    v8i gz8 = {0, 0, 0, 0, 0, 0, 0, 0};
    __builtin_amdgcn_tensor_load_to_lds(g0, g1, gz, gz, gz8, 0);
#else
    __builtin_amdgcn_tensor_load_to_lds(g0, g1, gz, gz, 0);
#endif
#if __has_builtin(__builtin_amdgcn_s_wait_tensorcnt)
    __builtin_amdgcn_s_wait_tensorcnt(0);
#else
    asm volatile("s_wait_tensorcnt 0x0" ::: "memory");
#endif
  }
  __syncthreads();
  // Threshold in place: sw = (s > 0.01) ? s : -1
  for (int i = tid; i < Nn; i += bs) {
    float s = sm[i];
    sm[i] = (s > 0.01f) ? s : -1.0f;
  }
#else
  // Fallback: per-lane strided loads (uncoalesced but correct)
  for (int i = tid; i < Nn; i += bs) {
    float s = cls[((size_t)b * Nn + i) * Cn + c];
    sm[i] = (s > 0.01f) ? s : -1.0f;
  }
#endif
  __syncthreads();

  const float*  bb  = boxes + (size_t)b * Nn * 4;
  const float4* bb4 = (const float4*)bb;
  float* osc = ws_sc + (size_t)bc * MAXDET;
  int*   oix = ws_ix + (size_t)bc * MAXDET;

  for (int det = 0; det < MAXDET; ++det) {
    // ---- block-wide argmax, first-index tiebreak (== jnp.argmax) ----------
    float bv = -FLT_MAX;
    int   bi = 0x7FFFFFFF;
    for (int i = tid; i < Nn; i += bs) {
      float v = sm[i];
      if (v > bv) { bv = v; bi = i; }
    }
    for (int off = 16; off > 0; off >>= 1) {
      float ov = __shfl_xor(bv, off, 32);
      int   oi = __shfl_xor(bi, off, 32);
      if (ov > bv || (ov == bv && oi < bi)) { bv = ov; bi = oi; }
    }
    if (lane == 0) { red_v[wid] = bv; red_i[wid] = bi; }
    __syncthreads();
    if (wid == 0) {
      bv = red_v[lane];                 // exactly 32 waves
      bi = red_i[lane];
      for (int off = 16; off > 0; off >>= 1) {
        float ov = __shfl_xor(bv, off, 32);
        int   oi = __shfl_xor(bi, off, 32);
        if (ov > bv || (ov == bv && oi < bi)) { bv = ov; bi = oi; }
      }
      if (lane == 0) {
        s_best_v = bv;
        s_best_i = bi;
        if (bv > 0.0f) {
          float4 q = bb4[bi];
          s_box[0] = q.x; s_box[1] = q.y; s_box[2] = q.z; s_box[3] = q.w;
          s_area   = (q.z - q.x) * (q.w - q.y);
          osc[det] = bv;
          oix[det] = bi;
        }
      }
    }
    __syncthreads();

    float best = s_best_v;
    if (best <= 0.0f) {
      // Reference: once invalid, state frozen -> remaining slots are (-1,-1)
      if (tid == 0) {
        for (int d = det; d < MAXDET; ++d) { osc[d] = -1.0f; oix[d] = -1; }
      }
      break;                            // uniform across block
    }

    const int   besti = s_best_i;
    const float ax1 = s_box[0], ay1 = s_box[1], ax2 = s_box[2], ay2 = s_box[3];
    const float areaA = s_area;

    // ---- IoU suppression against all surviving candidates -----------------
    for (int i = tid; i < Nn; i += bs) {
      if (i + bs < Nn) __builtin_prefetch(&bb4[i + bs], 0, 1);
      if (sm[i] < 0.0f) continue;       // dead entries stay dead
      if (i == besti) { sm[i] = -1.0f; continue; }
      float4 q = bb4[i];
      float xx1 = fmaxf(ax1, q.x);
      float yy1 = fmaxf(ay1, q.y);
      float xx2 = fminf(ax2, q.z);
      float yy2 = fminf(ay2, q.w);
      float inter = fmaxf(xx2 - xx1, 0.0f) * fmaxf(yy2 - yy1, 0.0f);
      float areaB = (q.z - q.x) * (q.w - q.y);
      float iou   = inter / (areaA + areaB - inter + 1e-8f);
      if (iou > 0.5f) sm[i] = -1.0f;
    }
    __syncthreads();
  }
}

// ---------------------------------------------------------------------------
// Kernel 2: per-image top-k over the C*MAXDET = 3000 (score, anchor) pairs,
// iterative argmax with removal == stable jax.lax.top_k, then gather+pad.
// Output tuple concatenated flat: boxes[B,100,4] | scores[B,100] |
// labels[B,100] | rot[B,100,3] | trans[B,100,3]  (9600 floats).
// ---------------------------------------------------------------------------
__global__ __launch_bounds__(256, 1)
void topk_kernel(const float* __restrict__ ws_sc, const int* __restrict__ ws_ix,
                 const float* __restrict__ boxes, const float* __restrict__ rot,
                 const float* __restrict__ trans, float* __restrict__ out) {
  __shared__ float sc[Cn * MAXDET];
  __shared__ int   an[Cn * MAXDET];
  __shared__ float red_v[8];
  __shared__ int   red_i[8];

  const int b    = blockIdx.x;
  const int tid  = threadIdx.x;
  const int bs   = blockDim.x;         // 256 = 8 waves
  const int lane = tid & 31;
  const int wid  = tid >> 5;

  for (int i = tid; i < Cn * MAXDET; i += bs) {
    sc[i] = ws_sc[(size_t)b * Cn * MAXDET + i];
    an[i] = ws_ix[(size_t)b * Cn * MAXDET + i];
  }
  __syncthreads();

  for (int k = 0; k < MAXDET; ++k) {
    float bv = -FLT_MAX;
    int   bi = 0x7FFFFFFF;
    for (int i = tid; i < Cn * MAXDET; i += bs) {
      float v = sc[i];
      if (v > bv) { bv = v; bi = i; }
    }
    for (int off = 16; off > 0; off >>= 1) {
      float ov = __shfl_xor(bv, off, 32);
      int   oi = __shfl_xor(bi, off, 32);
      if (ov > bv || (ov == bv && oi < bi)) { bv = ov; bi = oi; }
    }
    if (lane == 0) { red_v[wid] = bv; red_i[wid] = bi; }
    __syncthreads();
    if (tid == 0) {
      bv = red_v[0]; bi = red_i[0];
      for (int w = 1; w < 8; ++w) {
        if (red_v[w] > bv || (red_v[w] == bv && red_i[w] < bi)) {
          bv = red_v[w]; bi = red_i[w];
        }
      }
      const int   pos    = bi;
      const int   label  = pos / MAXDET;
      const int   anchor = an[pos];
      const bool  valid  = anchor >= 0;
      const int   safe   = anchor > 0 ? anchor : 0;
      const int   o      = b * MAXDET + k;

      float4 q = ((const float4*)boxes)[(size_t)b * Nn + safe];
      out[(size_t)o * 4 + 0] = valid ? q.x : -1.0f;
      out[(size_t)o * 4 + 1] = valid ? q.y : -1.0f;
      out[(size_t)o * 4 + 2] = valid ? q.z : -1.0f;
      out[(size_t)o * 4 + 3] = valid ? q.w : -1.0f;
      out[3200 + o] = valid ? bv : -1.0f;
      out[4000 + o] = valid ? (float)label : -1.0f;
      const size_t src3 = ((size_t)b * Nn + safe) * 3;
      for (int j = 0; j < 3; ++j) {
        out[4800 + (size_t)o * 3 + j] = valid ? rot[src3 + j]   : -1.0f;
        out[7200 + (size_t)o * 3 + j] = valid ? trans[src3 + j] : -1.0f;
      }
      sc[pos] = -FLT_MAX;               // remove -> next argmax = next top-k
    }
    __syncthreads();
  }
}

extern "C" void kernel_launch(void* const* d_in, const int* in_sizes, int n_in,
                              void* d_out, int out_size, void* d_ws, size_t ws_size,
                              hipStream_t stream) {
  (void)in_sizes; (void)n_in; (void)out_size; (void)ws_size;
  const float* boxes = (const float*)d_in[0];   // [B,N,4]
  const float* cls   = (const float*)d_in[1];   // [B,N,C]
  const float* rot   = (const float*)d_in[2];   // [B,N,3]
  const float* trans = (const float*)d_in[3];   // [B,N,3]
  float* out = (float*)d_out;                   // 9600 floats

  // Workspace: [B*C*MAXDET] scores (96 KB) + [B*C*MAXDET] anchors (96 KB)
  float* ws_sc = (float*)d_ws;
  int*   ws_ix = (int*)((char*)d_ws + (size_t)Bn * Cn * MAXDET * sizeof(float));

  // 196,416 B dynamic LDS per block — legal on CDNA5 (320 KB/WGP), but above
  // the 64 KB default cap, so raise the attribute (idempotent, capture-safe).
  const size_t shmem = (size_t)Nn * sizeof(float);
  hipFuncSetAttribute((const void*)nms_kernel,
                      hipFuncAttributeMaxDynamicSharedMemorySize, (int)shmem);

  nms_kernel<<<Bn * Cn, 1024, shmem, stream>>>(boxes, cls, ws_sc, ws_ix);
  topk_kernel<<<Bn, 256, 0, stream>>>(ws_sc, ws_ix, boxes, rot, trans, out);
}